// CustomScatter_8040178778596
// MI455X (gfx1250) — compile-verified
//
#include <hip/hip_runtime.h>
#include <hip/hip_bf16.h>
#include <stdint.h>

// Row geometry fixed by the reference: D = 64 bf16 = 128 bytes = exactly one
// 128B-aligned cache line per row. 8 lanes x b128 cover one row.
#define ROW_BYTES   128
#define ROW_SHIFT   7      // row * 128
#define SUB_SHIFT   4      // sub-slice * 16B
#define SLICES      4      // rows-worth of 16B slices handled per thread (MLP per wave)

typedef unsigned int u32x4 __attribute__((ext_vector_type(4)));

// ---------------------------------------------------------------------------
// Fill a buffer with a 32-bit pattern using non-temporal b128 stores
// (streaming write; do not pollute the 192MB L2).
// ---------------------------------------------------------------------------
__global__ void __launch_bounds__(256)
cs_fill_kernel(u32x4* __restrict__ p, long long n16, unsigned pat) {
  long long i      = (long long)blockIdx.x * blockDim.x + threadIdx.x;
  long long stride = (long long)gridDim.x * blockDim.x;
  u32x4 v = {pat, pat, pat, pat};
  for (; i < n16; i += stride)
    __builtin_nontemporal_store(v, p + i);
}

// ---------------------------------------------------------------------------
// Pass 1 (map path): winner map. map[idx[i]] = max over i  ("last update wins",
// deterministic across replays). map is 16MB -> L2 resident on MI455X.
// ---------------------------------------------------------------------------
__global__ void __launch_bounds__(256)
cs_build_map_kernel(const int* __restrict__ idx, int* __restrict__ map,
                    long long n) {
  long long i = (long long)blockIdx.x * blockDim.x + threadIdx.x;
  if (i < n) atomicMax(&map[idx[i]], (int)i);
}

// ---------------------------------------------------------------------------
// Pass 2 (map path): one sweep over ALL output rows. Winning rows gather their
// source row through the CDNA5 async global->LDS->global b128 path; untouched
// rows get non-temporal zero stores. Every output byte written exactly once.
// Slice s: out_row = s>>3, sub = s&7 (16B per lane).
// ---------------------------------------------------------------------------
__global__ void __launch_bounds__(256)
cs_gather_emit_kernel(const char* __restrict__ in, const int* __restrict__ map,
                      char* __restrict__ out, long long n_slices) {
  __shared__ char lds[256 * 16 * SLICES];
  const long long T = (long long)gridDim.x * blockDim.x;
  const long long g = (long long)blockIdx.x * blockDim.x + threadIdx.x;
  // Generic pointer to LDS: low 32 bits are the LDS byte offset the async
  // instructions expect in their LDS-address VGPR.
  const unsigned lbase =
      (unsigned)(uintptr_t)(&lds[0]) + (unsigned)threadIdx.x * (16u * SLICES);

  int m[SLICES];
#pragma unroll
  for (int j = 0; j < SLICES; ++j) {
    long long s = g + (long long)j * T;
    m[j] = -1;
    if (s < n_slices) {
      long long r  = s >> 3;
      int sub      = (int)(s & 7);
      m[j]         = map[r];                       // 8 lanes share one line
      if (m[j] >= 0) {
        unsigned long long gsrc = (unsigned long long)(uintptr_t)
            (in + ((long long)m[j] << ROW_SHIFT) + ((long long)sub << SUB_SHIFT));
        unsigned loff = lbase + (unsigned)(j * 16);
        asm volatile("global_load_async_to_lds_b128 %0, %1, off th:TH_LOAD_NT"
                     :: "v"(loff), "v"(gsrc) : "memory");
      }
    }
  }
  asm volatile("s_wait_asynccnt 0x0" ::: "memory");
#pragma unroll
  for (int j = 0; j < SLICES; ++j) {
    long long s = g + (long long)j * T;
    if (s < n_slices) {
      long long r  = s >> 3;
      int sub      = (int)(s & 7);
      char* gdst = out + (r << ROW_SHIFT) + ((long long)sub << SUB_SHIFT);
      if (m[j] >= 0) {
        unsigned loff = lbase + (unsigned)(j * 16);
        asm volatile("global_store_async_from_lds_b128 %0, %1, off th:TH_STORE_NT"
                     :: "v"((unsigned long long)(uintptr_t)gdst), "v"(loff)
                     : "memory");
      } else {
        u32x4 z = {0u, 0u, 0u, 0u};
        __builtin_nontemporal_store(z, (u32x4*)gdst);
      }
    }
  }
}

// ---------------------------------------------------------------------------
// Fallback scatter (no workspace): direct async copy input-row -> out[idx]
// (zero-fill done by cs_fill_kernel beforehand). Slice s: in_row = s>>3.
// ---------------------------------------------------------------------------
__global__ void __launch_bounds__(256)
cs_scatter_async_kernel(const char* __restrict__ in, const int* __restrict__ idx,
                        char* __restrict__ out, long long n_slices) {
  __shared__ char lds[256 * 16 * SLICES];
  const long long T = (long long)gridDim.x * blockDim.x;
  const long long g = (long long)blockIdx.x * blockDim.x + threadIdx.x;
  const unsigned lbase =
      (unsigned)(uintptr_t)(&lds[0]) + (unsigned)threadIdx.x * (16u * SLICES);

  int dst_row[SLICES];
#pragma unroll
  for (int j = 0; j < SLICES; ++j) {
    long long s = g + (long long)j * T;
    if (s < n_slices) {
      long long r = s >> 3;
      int sub     = (int)(s & 7);
      unsigned long long gsrc = (unsigned long long)(uintptr_t)
          (in + (r << ROW_SHIFT) + ((long long)sub << SUB_SHIFT));
      unsigned loff = lbase + (unsigned)(j * 16);
      asm volatile("global_load_async_to_lds_b128 %0, %1, off th:TH_LOAD_NT"
                   :: "v"(loff), "v"(gsrc) : "memory");
      dst_row[j] = idx[r];
    }
  }
  asm volatile("s_wait_asynccnt 0x0" ::: "memory");
#pragma unroll
  for (int j = 0; j < SLICES; ++j) {
    long long s = g + (long long)j * T;
    if (s < n_slices) {
      int sub = (int)(s & 7);
      unsigned long long gdst = (unsigned long long)(uintptr_t)
          (out + ((long long)dst_row[j] << ROW_SHIFT) + ((long long)sub << SUB_SHIFT));
      unsigned loff = lbase + (unsigned)(j * 16);
      asm volatile("global_store_async_from_lds_b128 %0, %1, off th:TH_STORE_NT"
                   :: "v"(gdst), "v"(loff) : "memory");
    }
  }
}

// ---------------------------------------------------------------------------
extern "C" void kernel_launch(void* const* d_in, const int* in_sizes, int n_in,
                              void* d_out, int out_size, void* d_ws, size_t ws_size,
                              hipStream_t stream) {
  const char* in  = (const char*)d_in[0];   // bf16 [N, 64]
  const int*  idx = (const int*)d_in[1];    // int  [N]
  char*       out = (char*)d_out;           // bf16 [OUT, 64]

  const long long n_in_elems = in_sizes[0];           // N * D
  const long long N          = in_sizes[1];
  const long long D          = n_in_elems / N;        // 64
  const long long OUT        = (long long)out_size / D;
  const long long slices_per_row = (D * 2) / 16;      // 8

  const int TPB = 256;

  if (ws_size >= (size_t)OUT * sizeof(int) && D == 64) {
    // ---- deterministic map path (~0.79 GB of traffic -> ~34us at 23.3 TB/s)
    int* map = (int*)d_ws;

    long long map16 = (OUT * sizeof(int)) / 16;
    cs_fill_kernel<<<4096, TPB, 0, stream>>>((u32x4*)map, map16, 0xFFFFFFFFu);

    long long bblocks = (N + TPB - 1) / TPB;
    cs_build_map_kernel<<<(unsigned)bblocks, TPB, 0, stream>>>(idx, map, N);

    long long n_slices = OUT * slices_per_row;        // 33,554,432
    long long per_blk  = (long long)TPB * SLICES;
    long long gblocks  = (n_slices + per_blk - 1) / per_blk;   // 32768
    cs_gather_emit_kernel<<<(unsigned)gblocks, TPB, 0, stream>>>(in, map, out,
                                                                 n_slices);
  } else {
    // ---- fallback: zero-fill then direct async scatter (~1.03 GB -> ~44us)
    long long out16 = ((long long)out_size * 2) / 16;
    cs_fill_kernel<<<16384, TPB, 0, stream>>>((u32x4*)out, out16, 0u);

    long long n_slices = N * slices_per_row;          // 16,777,216
    long long per_blk  = (long long)TPB * SLICES;
    long long sblocks  = (n_slices + per_blk - 1) / per_blk;   // 16384
    cs_scatter_async_kernel<<<(unsigned)sblocks, TPB, 0, stream>>>(in, idx, out,
                                                                   n_slices);
  }
}